// SSIM3DLoss_18915035972082
// MI455X (gfx1250) — compile-verified
//
#include <hip/hip_runtime.h>

// ---------------------------------------------------------------------------
// Fused 3D-SSIM on gfx1250 (MI455X).
//  * W and H conv axes on the fp32 matrix pipe (V_WMMA_F32_16X16X4_F32,
//    exact fp32 MACs); D axis + SSIM math on VALU (co-executes with XDL).
//  * Double-buffered input planes prefetched by the Tensor Data Mover
//    (tensor_load_to_lds, TENSORcnt) so global->LDS traffic overlaps compute.
//  * Wave roles scalarized via readfirstlane -> pure SALU dispatch.
// ---------------------------------------------------------------------------

typedef float v2f __attribute__((ext_vector_type(2)));
typedef float v8f __attribute__((ext_vector_type(8)));
typedef unsigned int v4u __attribute__((ext_vector_type(4)));
typedef int v4i __attribute__((ext_vector_type(4)));
typedef int v8i __attribute__((ext_vector_type(8)));

#define NBATCH 4
#define D_DIM  64
#define H_DIM  256
#define W_DIM  256

#define TD_H   32            // output tile height  (H)
#define TD_W   16            // output tile width   (W)
#define ROWS   42            // TD_H + 10 halo rows
#define ROWSP  44            // padded rows in LDS (rows 42..43 stay zero)
#define XCOLS  26            // TD_W + 10 halo cols
#define XP     26            // plane stride = tile width (TDM-contiguous rows)
#define PLSZ   (ROWSP * XP + 2)  // +2: cols 26..27 of last row read as zeros
#define QROWS  44            // W-blurred plane rows (42 real + 2 zero)
#define QP     19            // Q row stride (bank-disjoint for m / m+8 stores)
#define BP     19            // BZ row stride
#define NF     5             // fields: x, y, x*x, y*y, x*y

#define C1v       1.0e-4f
#define C2v       9.0e-4f
#define SSIM_EPS  1.0e-8f

__device__ __forceinline__ float ssim_term(float a0, float a1, float a2,
                                           float a3, float a4) {
    float mux2 = a0 * a0, muy2 = a1 * a1, muxy = a0 * a1;
    float sx = a2 - mux2, sy = a3 - muy2, sxy = a4 - muxy;
    float num = (2.f * muxy + C1v) * (2.f * sxy + C2v);
    float den = (mux2 + muy2 + C1v) * (sx + sy + C2v);
    return num / (den + SSIM_EPS);
}

// TDM 2-D tile load: `nrows` rows of 26 fp32 from a 256-wide plane into
// contiguous LDS rows.  D# per CDNA5 ISA §8.3/8.4 (count=1, type=2,
// data_size=4B, tensor dims huge so no OOB path is exercised).
// This toolchain declares the 6-arg builtin: (v4u, v8i, v4i, v4i, v8i, i32);
// the trailing groups are unused for a 2-D tensor -> zero-filled.
__device__ __forceinline__ void tdm_load_2d(const float* gaddr,
                                            unsigned lds_byte_off, int nrows) {
    unsigned long long ga = (unsigned long long)(size_t)gaddr;
    v4u g0;
    g0.x = 1u;                                   // count=1, user mode
    g0.y = lds_byte_off;                         // lds_addr
    g0.z = (unsigned)ga;                         // global_addr[31:0]
    g0.w = (unsigned)((ga >> 32) & 0x01FFFFFFull) | (2u << 30); // addr hi | type=2
    v8i g1;
    g1[0] = 0x00020000;                          // data_size = 4 bytes
    g1[1] = (int)0xFFFF0000u;                    // tensor_dim0 = 0x7FFFFFFF (lo)
    g1[2] = (int)0xFFFF7FFFu;                    // dim0 hi | tensor_dim1 lo
    g1[3] = (int)((XCOLS << 16) | 0x7FFF);       // dim1 hi | tile_dim0 = 26
    g1[4] = nrows & 0xFFFF;                      // tile_dim1 (tile_dim2 = 0)
    g1[5] = W_DIM;                               // tensor_dim0_stride = 256
    g1[6] = 0;
    g1[7] = 0;
    v4i zz4 = {0, 0, 0, 0};
    v8i zz8 = {0, 0, 0, 0, 0, 0, 0, 0};
    __builtin_amdgcn_tensor_load_to_lds(g0, g1, zz4, zz4, zz8, 0);
}

// W-axis conv for one field: out(16x16) = A(data 16x28) x B(gauss band 28x16).
// F: 0=x, 1=y, 2=x*x, 3=y*y, 4=x*y.  Instantiated under a *scalar* switch.
// Fragments are preloaded so DS latency of all 7 chunks overlaps.
template <int F>
__device__ __forceinline__ v8f wconv_run(const float* __restrict__ LXc,
                                         const float* __restrict__ LYc,
                                         int rowbase, int hi,
                                         const float (&bwv)[7][2]) {
    v2f a[7];
    #pragma unroll
    for (int q = 0; q < 7; ++q) {
        const int o = rowbase + 4 * q + 2 * hi;
        if constexpr (F == 0) {
            a[q].x = LXc[o]; a[q].y = LXc[o + 1];
        } else if constexpr (F == 1) {
            a[q].x = LYc[o]; a[q].y = LYc[o + 1];
        } else if constexpr (F == 2) {
            float x0 = LXc[o], x1 = LXc[o + 1];
            a[q].x = x0 * x0; a[q].y = x1 * x1;
        } else if constexpr (F == 3) {
            float y0 = LYc[o], y1 = LYc[o + 1];
            a[q].x = y0 * y0; a[q].y = y1 * y1;
        } else {
            float x0 = LXc[o], x1 = LXc[o + 1];
            float y0 = LYc[o], y1 = LYc[o + 1];
            a[q].x = x0 * y0; a[q].y = x1 * y1;
        }
    }
    v8f c = {};
    #pragma unroll
    for (int q = 0; q < 7; ++q) {
        v2f bb; bb.x = bwv[q][0]; bb.y = bwv[q][1];
        c = __builtin_amdgcn_wmma_f32_16x16x4_f32(
                false, a[q], false, bb, (short)0, c, false, false);
    }
    return c;
}

__global__ __launch_bounds__(512)
void ssim3d_fused(const float* __restrict__ X, const float* __restrict__ Y,
                  const float* __restrict__ K3, float* __restrict__ part)
{
    __shared__ float PL[2][2][PLSZ];        // [parity][x/y] haloed planes
    __shared__ float QS[NF * QROWS * QP];   // W-blurred planes (5 fields)
    __shared__ float BZ[NF * TD_H * BP];    // HW-blurred planes (5 fields)
    __shared__ float G1[16];                // separable 1-D Gaussian

    const int tid  = threadIdx.x;
    const int lane = tid & 31;
    const int hi   = lane >> 4;     // half-wave (K offset in WMMA frags)
    const int l16  = lane & 15;

    // Wave id in an SGPR: role dispatch is scalar control flow, EXEC stays
    // all-ones around WMMA.
    const int wave = __builtin_amdgcn_readfirstlane(tid >> 5);

    // 1-D kernel = marginal sums of the (normalized, separable) 3-D kernel.
    if (tid < 11) {
        float s = 0.f;
        for (int j = 0; j < 121; ++j) s += K3[tid * 121 + j];
        G1[tid] = s;
    }
    for (int i = tid; i < 2 * 2 * PLSZ; i += 512) ((float*)PL)[i] = 0.f;
    __syncthreads();

    float g1r[11];
    #pragma unroll
    for (int t = 0; t < 11; ++t) g1r[t] = G1[t];

    auto gtab = [&](int i) -> float { return ((unsigned)i <= 10u) ? G1[i] : 0.f; };

    // W-conv coefficient B fragments (band matrix, same for all W waves).
    // B 4x16 f32 layout: VGPR v, lane: K = v + 2*hi, N = lane%16.
    float bwv[7][2];
    #pragma unroll
    for (int q = 0; q < 7; ++q) {
        bwv[q][0] = gtab(4 * q + 2 * hi     - l16);
        bwv[q][1] = gtab(4 * q + 2 * hi + 1 - l16);
    }
    // H-conv coefficient A fragments (band matrix, row-tile dependent).
    // A 16x4 f32 layout: VGPR v, lane: M = lane%16, K = v + 2*hi.
    const int rtH = wave & 1;
    const int rb  = rtH * 16;
    const int qg0 = rtH ? 4 : 0;
    float ahv[7][2];
    #pragma unroll
    for (int q = 0; q < 7; ++q) {
        int kk = 4 * (qg0 + q) + 2 * hi;
        ahv[q][0] = gtab(kk     - rb - l16);
        ahv[q][1] = gtab(kk + 1 - rb - l16);
    }

    const int bx  = blockIdx.x;          // 512 blocks = 4 batches * 8 * 16 tiles
    const int b   = bx >> 7;
    const int rem = bx & 127;
    const int h0  = (rem >> 4) * TD_H;
    const int w0  = (rem & 15) * TD_W;

    const int fW  = wave / 3;            // scalar W-conv role
    const int rtW = wave - fW * 3;
    const int fH  = wave >> 1;           // scalar H-conv role

    const int myh = tid >> 4;
    const int myw = tid & 15;

    // TDM eligibility: W-halo fully inside the image (stride stays 26);
    // top/bottom rows are clipped via tile_dim1, untouched rows stay zero.
    const bool tdm_ok = (w0 >= 16) && (w0 <= 224);
    const int  r0c    = (h0 == 0) ? 5 : 0;                  // first valid row
    const int  rend   = (h0 == 224) ? 37 : 42;              // one past last
    const int  nrows  = rend - r0c;

    auto load_plane_manual = [&](int z, int p) {
        for (int i = tid; i < ROWS * XCOLS; i += 512) {
            int r = i / XCOLS;
            int c = i - r * XCOLS;
            int hh = h0 - 5 + r;
            int ww = w0 - 5 + c;
            float xv = 0.f, yv = 0.f;
            if (hh >= 0 && hh < H_DIM && ww >= 0 && ww < W_DIM) {
                int idx = ((b * D_DIM + z) * H_DIM + hh) * W_DIM + ww;
                xv = X[idx]; yv = Y[idx];
            }
            PL[p][0][r * XP + c] = xv;
            PL[p][1][r * XP + c] = yv;
        }
    };
    auto issue_load = [&](int z, int p) {
        if (tdm_ok) {
            if (wave == 0) {                       // one DMA pair per block
                int idx = ((b * D_DIM + z) * H_DIM + (h0 - 5 + r0c)) * W_DIM
                          + (w0 - 5);
                unsigned lx = (unsigned)(size_t)(void*)&PL[p][0][0]
                              + (unsigned)(r0c * XP * 4);
                unsigned ly = (unsigned)(size_t)(void*)&PL[p][1][0]
                              + (unsigned)(r0c * XP * 4);
                tdm_load_2d(X + idx, lx, nrows);
                tdm_load_2d(Y + idx, ly, nrows);
            }
        } else {
            load_plane_manual(z, p);
        }
    };

    float acc[11][5];                    // depth-axis shift-register accumulators
    #pragma unroll
    for (int j = 0; j < 11; ++j)
        #pragma unroll
        for (int f = 0; f < 5; ++f) acc[j][f] = 0.f;
    float ssum = 0.f;

    // Prologue: plane 0 into parity 0.
    issue_load(0, 0);
    if (tdm_ok && wave == 0) __builtin_amdgcn_s_wait_tensorcnt(0);
    __syncthreads();

    for (int z = 0; z < D_DIM; ++z) {
        const int cur = z & 1;
        // Prefetch next plane into the other buffer (TDM: fully async).
        if (z + 1 < D_DIM) issue_load(z + 1, 1 - cur);

        const float* LXc = &PL[cur][0][0];
        const float* LYc = &PL[cur][1][0];

        // -------- W-axis conv via WMMA: 15 wave-tasks (5 fields x 3 row-tiles)
        if (wave < 15) {
            int row = rtW * 16 + l16;
            if (row > QROWS - 1) row = QROWS - 1;  // clamped rows read zeros
            const int rowbase = row * XP;
            v8f c;
            switch (fW) {                           // scalar switch
                case 0:  c = wconv_run<0>(LXc, LYc, rowbase, hi, bwv); break;
                case 1:  c = wconv_run<1>(LXc, LYc, rowbase, hi, bwv); break;
                case 2:  c = wconv_run<2>(LXc, LYc, rowbase, hi, bwv); break;
                case 3:  c = wconv_run<3>(LXc, LYc, rowbase, hi, bwv); break;
                default: c = wconv_run<4>(LXc, LYc, rowbase, hi, bwv); break;
            }
            // D 16x16 f32: VGPR j -> row j (lanes 0-15) / j+8 (lanes 16-31)
            #pragma unroll
            for (int j = 0; j < 8; ++j) {
                int orow = rtW * 16 + j + 8 * hi;
                if (orow < QROWS) QS[(fW * QROWS + orow) * QP + l16] = c[j];
            }
        }
        __syncthreads();

        // -------- H-axis conv via WMMA: 10 wave-tasks (5 fields x 2 row-tiles)
        if (wave < 10) {
            const int qbase = fH * QROWS * QP + l16;
            v2f bbv[7];
            #pragma unroll
            for (int q = 0; q < 7; ++q) {          // preload all B fragments
                int kk = 4 * (qg0 + q) + 2 * hi;
                bbv[q].x = QS[qbase + kk * QP];
                bbv[q].y = QS[qbase + (kk + 1) * QP];
            }
            v8f c = {};
            #pragma unroll
            for (int q = 0; q < 7; ++q) {
                v2f a;  a.x = ahv[q][0]; a.y = ahv[q][1];
                c = __builtin_amdgcn_wmma_f32_16x16x4_f32(
                        false, a, false, bbv[q], (short)0, c, false, false);
            }
            #pragma unroll
            for (int j = 0; j < 8; ++j) {
                int orow = rtH * 16 + j + 8 * hi;  // 0..31
                BZ[(fH * TD_H + orow) * BP + l16] = c[j];
            }
        }
        __syncthreads();

        // -------- depth-axis 11-tap conv (VALU, co-executes with WMMA) ------
        float bzv[5];
        #pragma unroll
        for (int f = 0; f < 5; ++f) bzv[f] = BZ[(f * TD_H + myh) * BP + myw];
        #pragma unroll
        for (int j = 0; j < 11; ++j) {             // acc[j] <-> out plane z-5+j
            float w = g1r[10 - j];
            #pragma unroll
            for (int f = 0; f < 5; ++f) acc[j][f] = fmaf(w, bzv[f], acc[j][f]);
        }
        if (z >= 5)                                 // out plane z-5 complete
            ssum += ssim_term(acc[0][0], acc[0][1], acc[0][2], acc[0][3], acc[0][4]);
        #pragma unroll
        for (int j = 0; j < 10; ++j)
            #pragma unroll
            for (int f = 0; f < 5; ++f) acc[j][f] = acc[j + 1][f];
        #pragma unroll
        for (int f = 0; f < 5; ++f) acc[10][f] = 0.f;

        // Next plane must be fully in LDS before anyone reads it.
        if (tdm_ok && wave == 0) __builtin_amdgcn_s_wait_tensorcnt(0);
        __syncthreads();
    }
    // Tail flush: planes 59..63 (missing contributions are zero-padded planes).
    #pragma unroll
    for (int e = 0; e < 5; ++e) {
        ssum += ssim_term(acc[0][0], acc[0][1], acc[0][2], acc[0][3], acc[0][4]);
        #pragma unroll
        for (int j = 0; j < 10; ++j)
            #pragma unroll
            for (int f = 0; f < 5; ++f) acc[j][f] = acc[j + 1][f];
    }

    // Deterministic fixed-order block reduction (reuse QS).
    __syncthreads();
    QS[tid] = ssum;
    __syncthreads();
    #pragma unroll
    for (int s = 256; s > 0; s >>= 1) {
        if (tid < s) QS[tid] += QS[tid + s];
        __syncthreads();
    }
    if (tid == 0) part[blockIdx.x] = QS[0];
}

__global__ __launch_bounds__(512)
void ssim3d_finalize(const float* __restrict__ part, float* __restrict__ out)
{
    __shared__ float red[512];
    int tid = threadIdx.x;
    red[tid] = part[tid];
    __syncthreads();
    #pragma unroll
    for (int s = 256; s > 0; s >>= 1) {
        if (tid < s) red[tid] += red[tid + s];
        __syncthreads();
    }
    if (tid == 0) out[0] = 1.0f - red[0] / 16777216.0f;   // 4*64*256*256
}

extern "C" void kernel_launch(void* const* d_in, const int* in_sizes, int n_in,
                              void* d_out, int out_size, void* d_ws, size_t ws_size,
                              hipStream_t stream) {
    (void)in_sizes; (void)n_in; (void)out_size; (void)ws_size;
    const float* x  = (const float*)d_in[0];
    const float* y  = (const float*)d_in[1];
    const float* k3 = (const float*)d_in[2];
    float* part = (float*)d_ws;                 // 512 block partials
    ssim3d_fused<<<dim3(512), dim3(512), 0, stream>>>(x, y, k3, part);
    ssim3d_finalize<<<dim3(1), dim3(512), 0, stream>>>(part, (float*)d_out);
}